// GAT_4028679324259
// MI455X (gfx1250) — compile-verified
//
#include <hip/hip_runtime.h>
#include <math.h>

typedef float v2f __attribute__((ext_vector_type(2)));
typedef float v8f __attribute__((ext_vector_type(8)));

#define NEG_SLOPE 0.2f

// float atomic max via int-max / uint-min ordering trick (valid with -inf init)
__device__ __forceinline__ void atomic_max_float(float* addr, float val) {
  if (val >= 0.0f) {
    atomicMax((int*)addr, __float_as_int(val));
  } else {
    atomicMin((unsigned int*)addr, __float_as_uint(val));
  }
}

// -------------------------------------------------------------------------
// Y[N,M] = X[N,K] @ W[M,K]^T  using V_WMMA_F32_16X16X4_F32 (fp32 matrix core)
// One wave computes one 16x16 output tile; K consumed 4 at a time.
// A layout (ISA 7.12.2, 32-bit A 16x4): lanes 0-15 row=lane, K={0,1};
// lanes 16-31 row=lane-16, K={2,3}.  B mirrors with lane = output column.
// -------------------------------------------------------------------------
__global__ void gat_gemm_wmma(const float* __restrict__ X,
                              const float* __restrict__ W,
                              float* __restrict__ Y,
                              int K, int M, int numTiles, int tilesM) {
  int tile = blockIdx.x * (blockDim.x >> 5) + (threadIdx.x >> 5);
  if (tile >= numTiles) return;               // wave-uniform: EXEC stays all-1s
  int tm = tile % tilesM;                     // column tile
  int tn = tile / tilesM;                     // row tile
  int lane = threadIdx.x & 31;
  int half = lane >> 4;                       // 0: K lo pair, 1: K hi pair
  int l    = lane & 15;

  const v2f* Xr = (const v2f*)(X + (size_t)(tn * 16 + l) * K);  // A row
  const v2f* Wr = (const v2f*)(W + (size_t)(tm * 16 + l) * K);  // B col (= W row)

  v8f c = {};
  for (int kk = 0; kk < K; kk += 4) {
    v2f a = Xr[(kk >> 1) + half];             // X[row, kk+2*half .. +1]
    v2f b = Wr[(kk >> 1) + half];             // W[col, kk+2*half .. +1]
    c = __builtin_amdgcn_wmma_f32_16x16x4_f32(false, a, false, b,
                                              (short)0, c, false, false);
  }
  // C/D layout: lane(l,half), VGPR v -> row = half*8 + v, col = l
  float* Yp = Y + (size_t)(tn * 16 + half * 8) * M + tm * 16 + l;
#pragma unroll
  for (int v = 0; v < 8; ++v) Yp[(size_t)v * M] = c[v];
}

// -------------------------------------------------------------------------
// Per node: el[n,h] = sum_f fsrc[n,h*64+f]*attn[h*64+f]; init emax/denom/rst.
// grid = N, blockDim = H*64
// -------------------------------------------------------------------------
__global__ void gat_node_prep(const float* __restrict__ fsrc,
                              const float* __restrict__ attn,
                              float* __restrict__ el, float* __restrict__ emax,
                              float* __restrict__ denom, float* __restrict__ rst,
                              int H) {
  int n = blockIdx.x;
  int t = threadIdx.x;
  int M = H << 6;                             // F = 64 always
  rst[(size_t)n * M + t] = 0.0f;
  if (t < H) {
    const float* fp = fsrc + (size_t)n * M + (t << 6);
    const float* ap = attn + (t << 6);
    float s = 0.0f;
#pragma unroll
    for (int f = 0; f < 64; ++f) s += fp[f] * ap[f];
    el[n * H + t]    = s;
    emax[n * H + t]  = -INFINITY;
    denom[n * H + t] = 0.0f;
  }
}

// e[edge,h] = leaky_relu(el[src,h]); segment max into emax[dst,h]
__global__ void gat_edge_score(const int* __restrict__ src,
                               const int* __restrict__ dst,
                               const float* __restrict__ el,
                               float* __restrict__ eedge,
                               float* __restrict__ emax,
                               int E, int H) {
  int t = blockIdx.x * blockDim.x + threadIdx.x;
  if (t >= E * H) return;
  int e = t / H, h = t - e * H;
  float v = el[src[e] * H + h];
  v = (v > 0.0f) ? v : NEG_SLOPE * v;
  eedge[t] = v;
  atomic_max_float(&emax[dst[e] * H + h], v);
}

// ee = exp(e - emax[dst]); denom[dst] += ee
__global__ void gat_edge_expsum(const int* __restrict__ dst,
                                float* __restrict__ eedge,
                                const float* __restrict__ emax,
                                float* __restrict__ denom,
                                int E, int H) {
  int t = blockIdx.x * blockDim.x + threadIdx.x;
  if (t >= E * H) return;
  int e = t / H, h = t - e * H;
  int d = dst[e];
  float v = expf(eedge[t] - emax[d * H + h]);
  eedge[t] = v;
  atomicAdd(&denom[d * H + h], v);
}

// rst[dst] += fsrc[src] * (ee/denom[dst]); one wave per edge, lanes over M
__global__ void gat_edge_aggregate(const int* __restrict__ src,
                                   const int* __restrict__ dst,
                                   const float* __restrict__ fsrc,
                                   const float* __restrict__ eedge,
                                   const float* __restrict__ denom,
                                   float* __restrict__ rst,
                                   int E, int H) {
  int e = (int)((blockIdx.x * (size_t)blockDim.x + threadIdx.x) >> 5);
  int lane = threadIdx.x & 31;
  if (e >= E) return;                         // wave-uniform exit
  int s = src[e], d = dst[e];
  int M = H << 6;
  for (int m = lane; m < M; m += 32) {
    int h = m >> 6;                           // F = 64
    float a = eedge[e * H + h] / denom[d * H + h];
    atomicAdd(&rst[(size_t)d * M + m], fsrc[(size_t)s * M + m] * a);
  }
}

// out = (relu?) (rst + bias[m])
__global__ void gat_bias_act(const float* __restrict__ rst,
                             const float* __restrict__ bias,
                             float* __restrict__ out,
                             int total, int M, int do_relu) {
  int t = blockIdx.x * blockDim.x + threadIdx.x;
  if (t >= total) return;
  int m = t % M;
  float v = rst[t] + bias[m];
  if (do_relu) v = fmaxf(v, 0.0f);
  out[t] = v;
}

// -------------------------------------------------------------------------
static void run_gat_layer(const float* X, int K, const float* W, int M,
                          const float* attn, int H,
                          const int* src, const int* dst, int N, int E,
                          float* fsrc, float* rst, float* el, float* emax,
                          float* denom, float* eedge, hipStream_t stream) {
  int tilesM = M >> 4, tilesN = N >> 4;
  int numTiles = tilesM * tilesN;
  gat_gemm_wmma<<<(numTiles + 7) / 8, 256, 0, stream>>>(X, W, fsrc, K, M,
                                                        numTiles, tilesM);
  gat_node_prep<<<N, M, 0, stream>>>(fsrc, attn, el, emax, denom, rst, H);
  int EH = E * H;
  gat_edge_score<<<(EH + 255) / 256, 256, 0, stream>>>(src, dst, el, eedge,
                                                       emax, E, H);
  gat_edge_expsum<<<(EH + 255) / 256, 256, 0, stream>>>(dst, eedge, emax,
                                                        denom, E, H);
  gat_edge_aggregate<<<(E + 7) / 8, 256, 0, stream>>>(src, dst, fsrc, eedge,
                                                      denom, rst, E, H);
}

extern "C" void kernel_launch(void* const* d_in, const int* in_sizes, int n_in,
                              void* d_out, int out_size, void* d_ws, size_t ws_size,
                              hipStream_t stream) {
  const float* feat  = (const float*)d_in[0];   // [N,128]
  const float* W0    = (const float*)d_in[1];   // [128,128]
  const float* attn0 = (const float*)d_in[2];   // [2,64]
  const float* bias0 = (const float*)d_in[3];   // [128]
  const float* W1    = (const float*)d_in[4];   // [128,128]
  const float* attn1 = (const float*)d_in[5];   // [2,64]
  const float* bias1 = (const float*)d_in[6];   // [128]
  const float* W2    = (const float*)d_in[7];   // [64,128]
  const float* attn2 = (const float*)d_in[8];   // [1,64]
  const float* bias2 = (const float*)d_in[9];   // [64]
  const int*   src   = (const int*)d_in[10];    // [E]
  const int*   dst   = (const int*)d_in[11];    // [E]

  const int N = in_sizes[0] / 128;              // 50000 (divisible by 16)
  const int E = in_sizes[10];                   // 800000

  float* ws    = (float*)d_ws;
  float* hbuf  = ws;                 // [N,128] layer activations
  float* fsrc  = hbuf  + (size_t)N * 128;       // [N,128]
  float* rst   = fsrc  + (size_t)N * 128;       // [N,128]
  float* el    = rst   + (size_t)N * 128;       // [N,2]
  float* emax  = el    + (size_t)N * 2;         // [N,2]
  float* denom = emax  + (size_t)N * 2;         // [N,2]
  float* eedge = denom + (size_t)N * 2;         // [E,2]

  // ---- layer 0: [N,128] -> [N,128], H=2 ----
  run_gat_layer(feat, 128, W0, 128, attn0, 2, src, dst, N, E,
                fsrc, rst, el, emax, denom, eedge, stream);
  gat_bias_act<<<(N * 128 + 255) / 256, 256, 0, stream>>>(rst, bias0, hbuf,
                                                          N * 128, 128, 1);
  // ---- layer 1: [N,128] -> [N,128], H=2 ----
  run_gat_layer(hbuf, 128, W1, 128, attn1, 2, src, dst, N, E,
                fsrc, rst, el, emax, denom, eedge, stream);
  gat_bias_act<<<(N * 128 + 255) / 256, 256, 0, stream>>>(rst, bias1, hbuf,
                                                          N * 128, 128, 1);
  // ---- layer 2: [N,128] -> [N,64], H=1 (head-mean is identity) ----
  run_gat_layer(hbuf, 128, W2, 64, attn2, 1, src, dst, N, E,
                fsrc, rst, el, emax, denom, eedge, stream);
  gat_bias_act<<<(N * 64 + 255) / 256, 256, 0, stream>>>(rst, bias2,
                                                         (float*)d_out,
                                                         N * 64, 64, 0);
}